// ParallelVarPatchEmbedWind_79834852098104
// MI455X (gfx1250) — compile-verified
//
#include <hip/hip_runtime.h>
#include <math.h>

#define B_N 4
#define V_N 8
#define H_N 128
#define W_N 256
#define GH_N 32
#define GW_N 64
#define L_N 2048
#define D_N 768
#define NSEC 16
#define PI_F 3.14159265358979323846f

typedef float v2f __attribute__((ext_vector_type(2)));
typedef float v8f __attribute__((ext_vector_type(8)));

// ---------------- Kernel 1: per-batch wind sector ----------------
__global__ void wind_sector_kernel(const float* __restrict__ x, int* __restrict__ sec) {
    const int b = blockIdx.x;
    const float* xu = x + (size_t)b * V_N * H_N * W_N;          // channel 0 (u)
    const float* xv = xu + (size_t)H_N * W_N;                   // channel 1 (v)
    float su = 0.f, sv = 0.f;
    for (int i = threadIdx.x; i < H_N * W_N; i += blockDim.x) {
        su += xu[i];
        sv += xv[i];
    }
    __shared__ float ru[256];
    __shared__ float rv[256];
    ru[threadIdx.x] = su;
    rv[threadIdx.x] = sv;
    __syncthreads();
    for (int o = 128; o > 0; o >>= 1) {
        if (threadIdx.x < o) {
            ru[threadIdx.x] += ru[threadIdx.x + o];
            rv[threadIdx.x] += rv[threadIdx.x + o];
        }
        __syncthreads();
    }
    if (threadIdx.x == 0) {
        // atan2(v_mean, u_mean) == atan2(sum_v, sum_u): same positive divisor
        float ang = atan2f(rv[0], ru[0]);
        int s = (int)floorf((ang + PI_F) * ((float)NSEC / (2.0f * PI_F)));
        s = s < 0 ? 0 : (s > NSEC - 1 ? NSEC - 1 : s);
        sec[b] = s;
    }
}

// ---------------- Kernel 2: stable rank of each patch under wind order ----
__global__ void wind_rank_kernel(const int* __restrict__ sec, int* __restrict__ rank) {
    const int b = blockIdx.y;
    const int l = blockIdx.x * blockDim.x + threadIdx.x;   // 0..L-1
    const int s = sec[b];
    const float theta = -PI_F + ((float)s + 0.5f) * (2.0f * PI_F / (float)NSEC);
    const float ct = cosf(theta);
    const float st = sinf(theta);
    const float cx = (float)(l & (GW_N - 1));   // column
    const float cy = (float)(l >> 6);           // row (gw = 64)
    const float score = ct * cx + st * cy;
    int r = 0;
    for (int i = 0; i < GH_N; ++i) {
        const float rowbase = st * (float)i;
        const int lrow = i * GW_N;
        for (int j = 0; j < GW_N; ++j) {
            const float sj = ct * (float)j + rowbase;
            const int lj = lrow + j;
            // stable argsort: earlier index wins ties
            r += (sj < score) || ((sj == score) && (lj < l));
        }
    }
    rank[b * L_N + l] = r;
}

// ---------------- Kernel 3: WMMA f32 patch-projection + bias + rank scatter
// grid: (D/128, L/16, B*V), block: 256 (8 waves); wave -> 16x16 output tile.
__global__ void __launch_bounds__(256)
wmma_patch_embed_kernel(const float* __restrict__ x,
                        const float* __restrict__ w,      // [V, D, 16]
                        const float* __restrict__ bias,   // [V, D]
                        const int*   __restrict__ rank,   // [B, L]
                        float*       __restrict__ out) {  // [B, V, L, D]
    const int bv = blockIdx.z;
    const int b = bv / V_N;
    const int v = bv % V_N;
    const int l0 = blockIdx.y << 4;                     // 16-row tile of patches
    const int wave = threadIdx.x >> 5;                  // 0..7
    const int lane = threadIdx.x & 31;
    const int d0 = (blockIdx.x << 7) + (wave << 4);     // 16-col tile of D

    __shared__ int sRank[16];
    if (threadIdx.x < 16) sRank[threadIdx.x] = rank[b * L_N + l0 + threadIdx.x];
    __syncthreads();

    const int hl = lane >> 4;      // half-wave: selects K-pair 2*hl
    const int mn = lane & 15;      // m (A rows) / n (B,C cols)

    // ---- A: patches[l0+mn][k], k = 4*j + 2*hl + {0,1}  (contiguous float2)
    // patch l -> gi = l/64, gj = l%64; element p = py*4+px at x[... gi*4+py .. gj*4+px]
    const int l = l0 + mn;
    const int gi = l >> 6;
    const int gj = l & 63;
    const float* xp = x + ((((size_t)b * V_N + v) * H_N + (size_t)(gi << 2)) * W_N) + (gj << 2);
    v2f a[4];
#pragma unroll
    for (int j = 0; j < 4; ++j) {
        const float* p2 = xp + j * W_N + (hl << 1);   // py=j, px=2*hl
        a[j].x = p2[0];
        a[j].y = p2[1];
    }

    // ---- B: Wm^T[k][d0+mn] = w[v][d0+mn][k], k = 4*j + 2*hl + {0,1} (contiguous float2)
    const float* wp = w + ((size_t)v * D_N + (size_t)(d0 + mn)) * 16 + (hl << 1);
    v2f bm[4];
#pragma unroll
    for (int j = 0; j < 4; ++j) {
        bm[j].x = wp[(j << 2) + 0];
        bm[j].y = wp[(j << 2) + 1];
    }

    // ---- D = A x B + C over K=16 via 4 chained 16x16x4 f32 WMMAs
    v8f c = {0.f, 0.f, 0.f, 0.f, 0.f, 0.f, 0.f, 0.f};
#pragma unroll
    for (int j = 0; j < 4; ++j) {
        c = __builtin_amdgcn_wmma_f32_16x16x4_f32(
                /*neg_a=*/false, a[j],
                /*neg_b=*/false, bm[j],
                /*c_mod=*/(short)0, c,
                /*reuse_a=*/false, /*reuse_b=*/false);
    }

    // ---- bias + scatter rows by wind rank
    const float bval = bias[v * D_N + d0 + mn];
    const size_t outBase = (((size_t)b * V_N + v) * L_N) * D_N;
#pragma unroll
    for (int r = 0; r < 8; ++r) {
        const int m = r + (hl << 3);                 // C VGPR r holds rows r and r+8
        const int ro = sRank[m];
        out[outBase + (size_t)ro * D_N + d0 + mn] = c[r] + bval;
    }
}

// ---------------- Host launcher ----------------
extern "C" void kernel_launch(void* const* d_in, const int* in_sizes, int n_in,
                              void* d_out, int out_size, void* d_ws, size_t ws_size,
                              hipStream_t stream) {
    const float* x    = (const float*)d_in[0];   // [B,V,H,W]
    const float* w    = (const float*)d_in[1];   // [V,D,4,4]
    const float* bias = (const float*)d_in[2];   // [V,D]
    float* out = (float*)d_out;                  // [B,V,L,D]

    int* sec  = (int*)d_ws;          // B ints (padded to 16)
    int* rank = sec + 16;            // B*L ints

    wind_sector_kernel<<<B_N, 256, 0, stream>>>(x, sec);
    wind_rank_kernel<<<dim3(L_N / 256, B_N), 256, 0, stream>>>(sec, rank);
    wmma_patch_embed_kernel<<<dim3(D_N / 128, L_N / 16, B_N * V_N), 256, 0, stream>>>(
        x, w, bias, rank, out);
}